// GNNPretrainModel_21431886807579
// MI455X (gfx1250) — compile-verified
//
#include <hip/hip_runtime.h>
#include <math.h>

// Problem constants (fixed by the reference)
constexpr int N_  = 32768;
constexpr int E_  = 524288;
constexpr int F_  = 64;
constexpr int D_  = 64;
constexpr int ED_ = 32;
constexpr int L_  = 6;
constexpr int H_  = 2;
constexpr int HC_ = H_ * D_;        // 128
constexpr int EN_ = E_ + N_;        // edges + self loops

typedef float v2f __attribute__((ext_vector_type(2)));
typedef float v8f __attribute__((ext_vector_type(8)));

// ---- monotonic float<->uint encoding for atomicMax on floats ----
__device__ __forceinline__ unsigned fenc(float f) {
    unsigned u = __float_as_uint(f);
    return (u & 0x80000000u) ? ~u : (u | 0x80000000u);
}
__device__ __forceinline__ float fdec(unsigned u) {
    return __uint_as_float((u & 0x80000000u) ? (u ^ 0x80000000u) : ~u);
}
// fenc(-inf) == 0x007FFFFF
constexpr unsigned NEG_INF_ENC = 0x007FFFFFu;

// ---------------------------------------------------------------------------
// Generic fill
// ---------------------------------------------------------------------------
__global__ void fill_f32(float* p, float v, int n) {
    int t = blockIdx.x * blockDim.x + threadIdx.x;
    if (t < n) p[t] = v;
}

// degree accumulation (counts only real edges, like the reference)
__global__ void deg_kernel(const int* __restrict__ dst, float* __restrict__ deg) {
    int e = blockIdx.x * blockDim.x + threadIdx.x;
    if (e < E_) atomicAdd(&deg[dst[e]], 1.0f);
}

// loop_attr += segment_sum(edge_attr over dst)
__global__ void loopattr_accum(const int* __restrict__ dst,
                               const float* __restrict__ edge_attr,
                               float* __restrict__ loop_attr) {
    int e = blockIdx.x * blockDim.x + threadIdx.x;
    if (e >= E_) return;
    int d = dst[e];
    const float* row = edge_attr + (size_t)e * ED_;
    float* out = loop_attr + (size_t)d * ED_;
    #pragma unroll
    for (int k = 0; k < ED_; ++k) atomicAdd(&out[k], row[k]);
}

__global__ void loopattr_div(float* __restrict__ loop_attr, const float* __restrict__ deg) {
    int t = blockIdx.x * blockDim.x + threadIdx.x;
    if (t >= N_ * ED_) return;
    loop_attr[t] /= fmaxf(deg[t >> 5], 1.0f);
}

// ---------------------------------------------------------------------------
// WMMA row-tile GEMM: C[nrows x NCOLS] = A[nrows x K] @ B[K x NCOLS] (+bias)
// One wave32 computes 16 rows x NCOLS. Uses V_WMMA_F32_16X16X4_F32.
// Fragment layout per CDNA5 ISA 7.12.2 (32-bit A 16x4; B/C row-striped).
// ---------------------------------------------------------------------------
template <int K, int NCOLS>
__global__ void gemm_rowtile(const float* __restrict__ A, const float* __restrict__ B,
                             const float* __restrict__ bias, float* __restrict__ C,
                             int nrows) {
    constexpr int NT = NCOLS / 16;
    int wave = blockIdx.x * (blockDim.x >> 5) + (threadIdx.x >> 5);
    int m0 = wave * 16;
    if (m0 >= nrows) return;
    int lane = threadIdx.x & 31;
    int half = lane >> 4;
    int r    = lane & 15;

    __builtin_prefetch(B, 0, 1);   // global_prefetch_b8: pull weights toward WGP

    v8f acc[NT];
    #pragma unroll
    for (int t = 0; t < NT; ++t) {
        float bv = bias ? bias[t * 16 + r] : 0.0f;
        #pragma unroll
        for (int j = 0; j < 8; ++j) acc[t][j] = bv;
    }

    const float* arow = A + (size_t)(m0 + r) * K;
    #pragma unroll
    for (int k = 0; k < K; k += 4) {
        int k0 = k + 2 * half;
        v2f a;
        a.x = arow[k0];
        a.y = arow[k0 + 1];
        #pragma unroll
        for (int t = 0; t < NT; ++t) {
            v2f b;
            b.x = B[(size_t)k0 * NCOLS + t * 16 + r];
            b.y = B[(size_t)(k0 + 1) * NCOLS + t * 16 + r];
            acc[t] = __builtin_amdgcn_wmma_f32_16x16x4_f32(
                false, a, false, b, (short)0, acc[t], false, false);
        }
    }

    #pragma unroll
    for (int t = 0; t < NT; ++t)
        #pragma unroll
        for (int j = 0; j < 8; ++j)
            C[(size_t)(m0 + j + 8 * half) * NCOLS + t * 16 + r] = acc[t][j];
}

// ---------------------------------------------------------------------------
// LayerNorm + manual affine: one wave per node (D=64, 2 elems/lane)
// ---------------------------------------------------------------------------
__global__ void ln_kernel(const float* __restrict__ h,
                          const float* __restrict__ g, const float* __restrict__ b,
                          const float* __restrict__ mg, const float* __restrict__ mb,
                          float* __restrict__ hm) {
    int node = blockIdx.x * (blockDim.x >> 5) + (threadIdx.x >> 5);
    if (node >= N_) return;
    int lane = threadIdx.x & 31;
    const float* row = h + (size_t)node * D_;
    float v0 = row[lane], v1 = row[lane + 32];
    float s = v0 + v1;
    #pragma unroll
    for (int m = 16; m >= 1; m >>= 1) s += __shfl_xor(s, m, 32);
    float mu = s * (1.0f / D_);
    float d0 = v0 - mu, d1 = v1 - mu;
    float q = d0 * d0 + d1 * d1;
    #pragma unroll
    for (int m = 16; m >= 1; m >>= 1) q += __shfl_xor(q, m, 32);
    float inv = rsqrtf(q * (1.0f / D_) + 1e-5f);
    float* orow = hm + (size_t)node * D_;
    orow[lane]      = mg[lane]      * (g[lane]      * d0 * inv + b[lane])      + mb[lane];
    orow[lane + 32] = mg[lane + 32] * (g[lane + 32] * d1 * inv + b[lane + 32]) + mb[lane + 32];
}

// a_s[n,h] / a_d[n,h] dot products with attention vectors
__global__ void asd_kernel(const float* __restrict__ xh,
                           const float* __restrict__ att_src, const float* __restrict__ att_dst,
                           float* __restrict__ a_s, float* __restrict__ a_d) {
    int t = blockIdx.x * blockDim.x + threadIdx.x;
    if (t >= N_ * H_) return;
    int n = t >> 1, hh = t & 1;
    const float* row = xh + (size_t)n * HC_ + hh * D_;
    const float* ws = att_src + hh * D_;
    const float* wd = att_dst + hh * D_;
    float s1 = 0.f, s2 = 0.f;
    #pragma unroll
    for (int c = 0; c < D_; ++c) { float x = row[c]; s1 += x * ws[c]; s2 += x * wd[c]; }
    a_s[t] = s1; a_d[t] = s2;
}

// v[h, k] = sum_c lin_edge_W[k, h*C+c] * att_edge[h, c]   (folds edge GEMM into a dot)
__global__ void vfold_kernel(const float* __restrict__ lin_edge_W,
                             const float* __restrict__ att_edge,
                             float* __restrict__ vvec) {
    int t = threadIdx.x;           // 0..63
    if (t >= H_ * ED_) return;
    int hh = t >> 5, k = t & 31;
    const float* wrow = lin_edge_W + (size_t)k * HC_ + hh * D_;
    const float* ae = att_edge + hh * D_;
    float s = 0.f;
    #pragma unroll
    for (int c = 0; c < D_; ++c) s += wrow[c] * ae[c];
    vvec[hh * ED_ + k] = s;
}

// per-layer init: out accumulator = 0; amax = -inf(enc); den = 0
__global__ void layer_init(float* __restrict__ outacc, unsigned* __restrict__ amax,
                           float* __restrict__ den) {
    int t = blockIdx.x * blockDim.x + threadIdx.x;
    if (t < N_ * HC_) outacc[t] = 0.f;
    if (t < N_ * H_) { amax[t] = NEG_INF_ENC; den[t] = 0.f; }
}

// pass 1: alpha = leaky_relu(a_s[src]+a_d[dst]+edge_row@v); segment max via atomicMax
__global__ void alpha_kernel(const int* __restrict__ src, const int* __restrict__ dst,
                             const float* __restrict__ edge_attr, const float* __restrict__ loop_attr,
                             const float* __restrict__ vvec,
                             const float* __restrict__ a_s, const float* __restrict__ a_d,
                             float* __restrict__ alpha, unsigned* __restrict__ amax) {
    int e = blockIdx.x * blockDim.x + threadIdx.x;
    if (e >= EN_) return;
    int s, d; const float* erow;
    if (e < E_) { s = src[e]; d = dst[e]; erow = edge_attr + (size_t)e * ED_; }
    else        { int n = e - E_; s = n; d = n; erow = loop_attr + (size_t)n * ED_; }
    float ae0 = 0.f, ae1 = 0.f;
    #pragma unroll
    for (int k = 0; k < ED_; ++k) { float ev = erow[k]; ae0 += ev * vvec[k]; ae1 += ev * vvec[ED_ + k]; }
    #pragma unroll
    for (int hh = 0; hh < H_; ++hh) {
        float a = a_s[s * H_ + hh] + a_d[d * H_ + hh] + (hh ? ae1 : ae0);
        a = (a > 0.f) ? a : 0.2f * a;            // leaky_relu(0.2)
        alpha[(size_t)e * H_ + hh] = a;
        atomicMax(&amax[d * H_ + hh], fenc(a));
    }
}

// pass 2: ex = exp(alpha - amax[dst]); den[dst] += ex
__global__ void exden_kernel(const int* __restrict__ dst,
                             const float* __restrict__ alpha, const unsigned* __restrict__ amax,
                             float* __restrict__ exv, float* __restrict__ den) {
    int e = blockIdx.x * blockDim.x + threadIdx.x;
    if (e >= EN_) return;
    int d = (e < E_) ? dst[e] : (e - E_);
    #pragma unroll
    for (int hh = 0; hh < H_; ++hh) {
        float m = fdec(amax[d * H_ + hh]);
        float ex = expf(alpha[(size_t)e * H_ + hh] - m);
        exv[(size_t)e * H_ + hh] = ex;
        atomicAdd(&den[d * H_ + hh], ex);
    }
}

// pass 3: out[dst] += xh[src] * (ex/den).  One wave per edge row, 4 floats/lane.
__global__ void scatter_kernel(const int* __restrict__ src, const int* __restrict__ dst,
                               const float* __restrict__ xh,
                               const float* __restrict__ exv, const float* __restrict__ den,
                               float* __restrict__ outacc) {
    long gid = (long)blockIdx.x * blockDim.x + threadIdx.x;
    int e = (int)(gid >> 5);
    if (e >= EN_) return;
    int lane = (int)(gid & 31);
    int s, d;
    if (e < E_) { s = src[e]; d = dst[e]; } else { s = e - E_; d = s; }
    int c0 = lane * 4;
    int hh = c0 >> 6;
    float w = exv[(size_t)e * H_ + hh] / (den[d * H_ + hh] + 1e-16f);
    const float4 xv = *(const float4*)(xh + (size_t)s * HC_ + c0);
    float* o = outacc + (size_t)d * HC_ + c0;
    atomicAdd(&o[0], xv.x * w);
    atomicAdd(&o[1], xv.y * w);
    atomicAdd(&o[2], xv.z * w);
    atomicAdd(&o[3], xv.w * w);
}

// node update: h = relu(mean over heads + bias)
__global__ void node_update(const float* __restrict__ outacc, const float* __restrict__ gbias,
                            float* __restrict__ h) {
    int t = blockIdx.x * blockDim.x + threadIdx.x;
    if (t >= N_ * D_) return;
    int n = t >> 6, c = t & 63;
    float v = 0.5f * (outacc[(size_t)n * HC_ + c] + outacc[(size_t)n * HC_ + D_ + c]) + gbias[c];
    h[t] = fmaxf(v, 0.f);
}

// ---------------------------------------------------------------------------
// Edge head: out[e] = W2 . relu([h[src],h[dst],edge_attr] @ W1 + b1) + b2
// One wave per 16 edges; WMMA f32 16x16x4 over K=160, NCOLS=64 fused gather.
// ---------------------------------------------------------------------------
__global__ void edge_head(const int* __restrict__ src, const int* __restrict__ dst,
                          const float* __restrict__ h, const float* __restrict__ edge_attr,
                          const float* __restrict__ W1, const float* __restrict__ b1,
                          const float* __restrict__ W2, const float* __restrict__ b2,
                          float* __restrict__ out) {
    constexpr int K = 2 * D_ + ED_;     // 160
    constexpr int NT = 4;               // 64 cols
    int wave = blockIdx.x * (blockDim.x >> 5) + (threadIdx.x >> 5);
    int m0 = wave * 16;
    if (m0 >= E_) return;
    int lane = threadIdx.x & 31;
    int half = lane >> 4;
    int r    = lane & 15;

    int er = m0 + r;
    int es = src[er], ed = dst[er];
    const float* hs = h + (size_t)es * D_;
    const float* hd = h + (size_t)ed * D_;
    const float* ea = edge_attr + (size_t)er * ED_;

    v8f acc[NT];
    #pragma unroll
    for (int t = 0; t < NT; ++t)
        #pragma unroll
        for (int j = 0; j < 8; ++j) acc[t][j] = 0.f;

    #pragma unroll
    for (int k = 0; k < K; k += 4) {
        int k0 = k + 2 * half;
        const float* base;
        int kk;
        if (k < D_)            { base = hs; kk = k0; }
        else if (k < 2 * D_)   { base = hd; kk = k0 - D_; }
        else                   { base = ea; kk = k0 - 2 * D_; }
        v2f a;
        a.x = base[kk];
        a.y = base[kk + 1];
        #pragma unroll
        for (int t = 0; t < NT; ++t) {
            v2f b;
            b.x = W1[(size_t)k0 * D_ + t * 16 + r];
            b.y = W1[(size_t)(k0 + 1) * D_ + t * 16 + r];
            acc[t] = __builtin_amdgcn_wmma_f32_16x16x4_f32(
                false, a, false, b, (short)0, acc[t], false, false);
        }
    }

    // ReLU(+b1), multiply by W2 column weights, reduce across 64 columns
    float part[8];
    #pragma unroll
    for (int j = 0; j < 8; ++j) part[j] = 0.f;
    #pragma unroll
    for (int t = 0; t < NT; ++t) {
        int col = t * 16 + r;
        float b1v = b1[col];
        float w2v = W2[col];
        #pragma unroll
        for (int j = 0; j < 8; ++j) part[j] += fmaxf(acc[t][j] + b1v, 0.f) * w2v;
    }
    #pragma unroll
    for (int m = 1; m < 16; m <<= 1)
        #pragma unroll
        for (int j = 0; j < 8; ++j) part[j] += __shfl_xor(part[j], m, 16);
    if (r == 0) {
        float bb = b2[0];
        #pragma unroll
        for (int j = 0; j < 8; ++j) out[m0 + j + 8 * half] = part[j] + bb;
    }
}

// ---------------------------------------------------------------------------
extern "C" void kernel_launch(void* const* d_in, const int* in_sizes, int n_in,
                              void* d_out, int out_size, void* d_ws, size_t ws_size,
                              hipStream_t stream) {
    (void)in_sizes; (void)n_in; (void)out_size; (void)ws_size;

    const float* x        = (const float*)d_in[0];
    const int*   eidx     = (const int*)  d_in[1];
    const float* eattr    = (const float*)d_in[2];
    const float* mgamma   = (const float*)d_in[3];
    const float* mbeta    = (const float*)d_in[4];
    const float* embW     = (const float*)d_in[5];
    const float* embB     = (const float*)d_in[6];
    const float* lng      = (const float*)d_in[7];
    const float* lnb      = (const float*)d_in[8];
    const float* linW     = (const float*)d_in[9];
    const float* linEW    = (const float*)d_in[10];
    const float* attS     = (const float*)d_in[11];
    const float* attD     = (const float*)d_in[12];
    const float* attE     = (const float*)d_in[13];
    const float* gbias    = (const float*)d_in[14];
    const float* hW1      = (const float*)d_in[15];
    const float* hb1      = (const float*)d_in[16];
    const float* hW2      = (const float*)d_in[17];
    const float* hb2      = (const float*)d_in[18];
    float* out = (float*)d_out;

    const int* src = eidx;
    const int* dst = eidx + E_;

    // workspace carve-out (floats) — ~65 MB total
    float* W = (float*)d_ws;
    float* h       = W;  W += (size_t)N_ * D_;
    float* hm      = W;  W += (size_t)N_ * D_;
    float* xh      = W;  W += (size_t)N_ * HC_;
    float* outacc  = W;  W += (size_t)N_ * HC_;
    float* a_s     = W;  W += (size_t)N_ * H_;
    float* a_d     = W;  W += (size_t)N_ * H_;
    float* deg     = W;  W += (size_t)N_;
    float* lattr   = W;  W += (size_t)N_ * ED_;
    float* alpha   = W;  W += (size_t)EN_ * H_;
    float* exv     = W;  W += (size_t)EN_ * H_;
    float* den     = W;  W += (size_t)N_ * H_;
    float* vvec    = W;  W += 64;
    unsigned* amax = (unsigned*)W;  W += (size_t)N_ * H_;

    const int TB = 256;
    #define GRID(n) dim3(((n) + TB - 1) / TB)

    // ---- self-loop edge attr (mean of incoming edge_attr) ----
    fill_f32<<<GRID(N_), TB, 0, stream>>>(deg, 0.f, N_);
    fill_f32<<<GRID(N_ * ED_), TB, 0, stream>>>(lattr, 0.f, N_ * ED_);
    deg_kernel<<<GRID(E_), TB, 0, stream>>>(dst, deg);
    loopattr_accum<<<GRID(E_), TB, 0, stream>>>(dst, eattr, lattr);
    loopattr_div<<<GRID(N_ * ED_), TB, 0, stream>>>(lattr, deg);

    // ---- node embedding GEMM (WMMA) ----
    gemm_rowtile<F_, D_><<<dim3(N_ / 16 / 8), TB, 0, stream>>>(x, embW, embB, h, N_);

    // ---- GAT layers ----
    for (int l = 0; l < L_; ++l) {
        ln_kernel<<<dim3(N_ / 8), TB, 0, stream>>>(h, lng + l * D_, lnb + l * D_,
                                                   mgamma + l * D_, mbeta + l * D_, hm);
        gemm_rowtile<D_, HC_><<<dim3(N_ / 16 / 8), TB, 0, stream>>>(
            hm, linW + (size_t)l * D_ * HC_, nullptr, xh, N_);
        asd_kernel<<<GRID(N_ * H_), TB, 0, stream>>>(xh, attS + l * HC_, attD + l * HC_, a_s, a_d);
        vfold_kernel<<<dim3(1), dim3(64), 0, stream>>>(linEW + (size_t)l * ED_ * HC_,
                                                       attE + l * HC_, vvec);
        layer_init<<<GRID(N_ * HC_), TB, 0, stream>>>(outacc, amax, den);
        alpha_kernel<<<GRID(EN_), TB, 0, stream>>>(src, dst, eattr, lattr, vvec,
                                                   a_s, a_d, alpha, amax);
        exden_kernel<<<GRID(EN_), TB, 0, stream>>>(dst, alpha, amax, exv, den);
        scatter_kernel<<<dim3((unsigned)((size_t)EN_ * 32 / TB)), TB, 0, stream>>>(
            src, dst, xh, exv, den, outacc);
        node_update<<<GRID(N_ * D_), TB, 0, stream>>>(outacc, gbias + l * D_, h);
    }

    // ---- fused edge scoring head (gather + WMMA GEMM + ReLU + dot) ----
    edge_head<<<dim3(E_ / 16 / 8), TB, 0, stream>>>(src, dst, h, eattr,
                                                    hW1, hb1, hW2, hb2, out);
    #undef GRID
}